// TransformerQuantizerBak_57088705299030
// MI455X (gfx1250) — compile-verified
//
#include <hip/hip_runtime.h>
#include <stdint.h>

typedef __attribute__((ext_vector_type(2))) float v2f;
typedef __attribute__((ext_vector_type(8))) float v8f;

#define Gq  4
#define Kc  256
#define CSc 64
#define NB  16
#define Cc  256
#define HWc 4096

__device__ __forceinline__ uint32_t rotl32(uint32_t x, int r){ return (x<<r)|(x>>(32-r)); }

// Threefry-2x32, 20 rounds (JAX schedule)
__device__ __forceinline__ void threefry2x32(uint32_t k0, uint32_t k1,
                                             uint32_t x0, uint32_t x1,
                                             uint32_t &o0, uint32_t &o1){
  const uint32_t ks2 = k0 ^ k1 ^ 0x1BD11BDAu;
  uint32_t v0 = x0 + k0, v1 = x1 + k1;
  const int ra[4] = {13,15,26,6};
  const int rb[4] = {17,29,16,24};
#pragma unroll
  for(int i=0;i<4;i++){ v0 += v1; v1 = rotl32(v1, ra[i]); v1 ^= v0; }
  v0 += k1;  v1 += ks2 + 1u;
#pragma unroll
  for(int i=0;i<4;i++){ v0 += v1; v1 = rotl32(v1, rb[i]); v1 ^= v0; }
  v0 += ks2; v1 += k0 + 2u;
#pragma unroll
  for(int i=0;i<4;i++){ v0 += v1; v1 = rotl32(v1, ra[i]); v1 ^= v0; }
  v0 += k0;  v1 += k1 + 3u;
#pragma unroll
  for(int i=0;i<4;i++){ v0 += v1; v1 = rotl32(v1, rb[i]); v1 ^= v0; }
  v0 += k1;  v1 += ks2 + 4u;
#pragma unroll
  for(int i=0;i<4;i++){ v0 += v1; v1 = rotl32(v1, ra[i]); v1 ^= v0; }
  v0 += ks2; v1 += k0 + 5u;
  o0 = v0; o1 = v1;
}

// gumbel noise for flat index into (G,HW,NB,K) tensor (2^26 elements, key(42))
__device__ __forceinline__ float gumbel_at(uint32_t idx){
  const uint32_t half = 1u<<25;
  uint32_t j = idx & (half-1u);
  uint32_t o0,o1;
  threefry2x32(0u, 42u, j, j + half, o0, o1);
  uint32_t bits = (idx < half) ? o0 : o1;
  float f = __uint_as_float((bits>>9) | 0x3f800000u) - 1.0f;
  float u = f * (1.0f - 1e-20f) + 1e-20f;
  return -__logf(-__logf(u));
}

// keyT[g][d][k] = sum_c codebooks[g,k,c]*Wk[g,d,c] + bk[g,d]
// value[g][k][d] = sum_c codebooks[g,k,c]*Wv[g,d,c] + bv[g,d]
__global__ __launch_bounds__(256) void prep_kv(
    const float* __restrict__ cb,
    const float* __restrict__ Wk, const float* __restrict__ bk,
    const float* __restrict__ Wv, const float* __restrict__ bv,
    float* __restrict__ keyT, float* __restrict__ value){
  int t = blockIdx.x*blockDim.x + threadIdx.x;   // 0 .. G*K*CS-1
  int d = t & 63; int k = (t>>6) & 255; int g = t>>14;
  const float* cbr = cb + ((size_t)g*Kc + k)*CSc;
  const float* wkr = Wk + ((size_t)g*CSc + d)*CSc;
  const float* wvr = Wv + ((size_t)g*CSc + d)*CSc;
  float sk = bk[g*CSc + d], sv = bv[g*CSc + d];
#pragma unroll 8
  for(int c=0;c<CSc;c++){ float x = cbr[c]; sk = fmaf(x, wkr[c], sk); sv = fmaf(x, wvr[c], sv); }
  keyT [((size_t)g*CSc + d)*Kc + k] = sk;
  value[((size_t)g*Kc  + k)*CSc + d] = sv;
}

// One wave32 per 16-row tile (16 consecutive spatial positions, fixed batch n).
__global__ __launch_bounds__(128) void tq_main(
    const float* __restrict__ latent,
    const float* __restrict__ Wq, const float* __restrict__ bq,
    const float* __restrict__ keyT, const float* __restrict__ value,
    float* __restrict__ outq, float* __restrict__ outl){
  __shared__ float qlds[4][16*CSc];
  __shared__ int   idxlds[4][16];

  const int w    = threadIdx.x >> 5;
  const int lane = threadIdx.x & 31;
  const int tile = blockIdx.x * 4 + w;          // 0..4095
  const int nn   = tile & 15;
  const int s0   = (tile >> 4) << 4;            // 16 consecutive s
  const int half = (lane < 16) ? 0 : 1;
  const int lm   = lane & 15;

  for(int g=0; g<Gq; ++g){
    // ---- A fragments of x (streamed once: non-temporal loads) ----
    v2f a[16];
#pragma unroll
    for(int kc=0;kc<16;kc++){
      int c = g*CSc + kc*4 + half*2;
      const float* p = latent + ((size_t)nn*Cc + c)*HWc + s0 + lm;
      a[kc].x = __builtin_nontemporal_load(p);
      a[kc].y = __builtin_nontemporal_load(p + HWc);
    }
    // ---- GEMM1: query = x * Wq^T + bq  -> LDS ----
#pragma unroll
    for(int dc=0;dc<4;dc++){
      int d = dc*16 + lm;
      float bias = bq[g*CSc + d];
      v8f acc;
#pragma unroll
      for(int j=0;j<8;j++) acc[j] = bias;
#pragma unroll
      for(int kc=0;kc<16;kc++){
        int c = kc*4 + half*2;
        const float* p = Wq + ((size_t)g*CSc + d)*CSc + c;
        v2f b; b.x = p[0]; b.y = p[1];
        acc = __builtin_amdgcn_wmma_f32_16x16x4_f32(false, a[kc], false, b,
                                                    (short)0, acc, false, false);
      }
#pragma unroll
      for(int j=0;j<8;j++)
        qlds[w][(j + half*8)*CSc + d] = acc[j];
    }
    __syncthreads();
    // ---- query as A fragments ----
    v2f aq[16];
#pragma unroll
    for(int kc=0;kc<16;kc++){
      int d = kc*4 + half*2;
      aq[kc].x = qlds[w][lm*CSc + d];
      aq[kc].y = qlds[w][lm*CSc + d + 1];
    }
    // ---- GEMM2 + logits out (streaming NT stores) + gumbel running argmax ----
    float bestv[8]; int bestk[8];
#pragma unroll
    for(int j=0;j<8;j++){ bestv[j] = -3.4e38f; bestk[j] = 0; }
    for(int kchunk=0;kchunk<16;kchunk++){
      int kk = kchunk*16 + lm;
      v8f acc = {};
#pragma unroll
      for(int kc=0;kc<16;kc++){
        int d = kc*4 + half*2;
        const float* p = keyT + ((size_t)g*CSc + d)*Kc + kk;
        v2f b; b.x = p[0]; b.y = p[Kc];
        acc = __builtin_amdgcn_wmma_f32_16x16x4_f32(false, aq[kc], false, b,
                                                    (short)0, acc, false, false);
      }
#pragma unroll
      for(int j=0;j<8;j++){
        float lg = acc[j] * (1.0f/16.0f);            // /sqrt(K), K=256
        int s = s0 + j + half*8;
        __builtin_nontemporal_store(lg,
            &outl[(((size_t)(g*NB + nn))*HWc + s)*Kc + kk]);
        uint32_t e = (((uint32_t)(g*HWc + s))*NB + (uint32_t)nn)*Kc + (uint32_t)kk;
        float z = lg + gumbel_at(e);
        if(z > bestv[j]){ bestv[j] = z; bestk[j] = kk; }
      }
    }
    // cross-lane argmax within each 16-lane half (wave32)
#pragma unroll
    for(int m=1;m<16;m<<=1){
#pragma unroll
      for(int j=0;j<8;j++){
        float ov = __shfl_xor(bestv[j], m, 32);
        int   ok = __shfl_xor(bestk[j], m, 32);
        if(ov > bestv[j] || (ov == bestv[j] && ok < bestk[j])){ bestv[j]=ov; bestk[j]=ok; }
      }
    }
    if(lm == 0){
#pragma unroll
      for(int j=0;j<8;j++) idxlds[w][j + half*8] = bestk[j];
    }
    __syncthreads();
    // ---- quantized = value[g, argmax, :] gather (coalesced along s, NT stores) ----
    {
      int m = lm, s = s0 + m;
      int vidx = idxlds[w][m];
      const float* vrow = value + ((size_t)g*Kc + vidx)*CSc;
      for(int d = half; d < CSc; d += 2)
        __builtin_nontemporal_store(vrow[d],
            &outq[(((size_t)nn*Cc) + g*CSc + d)*HWc + s]);
    }
    __syncthreads();
  }
}

extern "C" void kernel_launch(void* const* d_in, const int* in_sizes, int n_in,
                              void* d_out, int out_size, void* d_ws, size_t ws_size,
                              hipStream_t stream){
  (void)in_sizes; (void)n_in; (void)out_size; (void)ws_size;
  const float* latent = (const float*)d_in[0];
  /* d_in[1] = temp: forward outputs are invariant to t>0 (argmax only) */
  const float* cb = (const float*)d_in[2];
  const float* Wq = (const float*)d_in[3];
  const float* bq = (const float*)d_in[4];
  const float* Wk = (const float*)d_in[5];
  const float* bk = (const float*)d_in[6];
  const float* Wv = (const float*)d_in[7];
  const float* bv = (const float*)d_in[8];

  float* keyT  = (float*)d_ws;                 // [G][CS][K]
  float* value = keyT + (size_t)Gq*CSc*Kc;     // [G][K][CS]
  float* outq  = (float*)d_out;                // (n,c,h,w) = 16*256*4096
  float* outl  = outq + (size_t)NB*Cc*HWc;     // (G,n,h,w,K)

  prep_kv<<<256, 256, 0, stream>>>(cb, Wk, bk, Wv, bv, keyT, value);
  tq_main<<<1024, 128, 0, stream>>>(latent, Wq, bq, keyT, value, outq, outl);
}